// LocalVariation_6932077215839
// MI455X (gfx1250) — compile-verified
//
#include <hip/hip_runtime.h>
#include <stdint.h>

// LocalVariation on MI455X (gfx1250):
//   out[b,k,y,x] = x[b,y,x] - clamp_neighbor(x, y+i-2, x+j-2),  24 taps of 5x5.
// Store-bandwidth bound (~420 MB moved, ~0.1 GFLOP) -> strategy:
//   * async global->LDS staging of a haloed tile (ASYNCcnt path, CDNA5)
//   * replicate padding via per-lane clamped addresses on the async copy
//   * 24x reuse of each input element from LDS
//   * non-temporal float4 stores for the 403 MB output stream (preserve L2)

#define KSZ 5
#define PAD 2
#define H   512
#define W   512
#define BATCH 16
#define NK  24                 // KSZ*KSZ - 1
#define TH  16                 // tile rows per block
#define LW  (W + 2 * PAD)      // 516 floats per LDS row
#define LH  (TH + 2 * PAD)     // 20 LDS rows
#define NTHREADS 256

typedef float v4f __attribute__((ext_vector_type(4)));

__device__ __forceinline__ uint32_t lds_byte_addr(const void* p) {
    // Generic (flat) shared pointers carry the LDS byte offset in their low
    // 32 bits (ISA 10.2 aperture mapping); async-load VDST wants that offset.
    return (uint32_t)(uintptr_t)p;
}

__global__ __launch_bounds__(NTHREADS) void localvar_kernel(
    const float* __restrict__ x, float* __restrict__ out) {
    __shared__ float tile[LH * LW];  // 41280 bytes

    const int b   = blockIdx.x;      // batch image
    const int y0  = blockIdx.y * TH; // first output row of tile
    const int tid = threadIdx.x;

    const float* xb = x + (size_t)b * (H * W);

    // ---- Phase 1: async global->LDS stage with replicate-clamped addresses
    const int TOT = LH * LW;  // 10320 elements
    for (int idx = tid; idx < TOT; idx += NTHREADS) {
        const int r  = idx / LW;
        const int c  = idx - r * LW;
        int gy = y0 - PAD + r;
        gy = gy < 0 ? 0 : (gy > H - 1 ? H - 1 : gy);
        int gx = c - PAD;
        gx = gx < 0 ? 0 : (gx > W - 1 ? W - 1 : gx);
        const float* gaddr = xb + gy * W + gx;
        const uint32_t laddr = lds_byte_addr(&tile[idx]);
        asm volatile("global_load_async_to_lds_b32 %0, %1, off"
                     :
                     : "v"(laddr), "v"(gaddr)
                     : "memory");
    }
    asm volatile("s_wait_asynccnt 0" ::: "memory");
    __syncthreads();  // s_barrier_signal/s_barrier_wait on gfx1250

    // ---- Phase 2: 24 taps, float4 per thread-slot, NT B128 stores
    const size_t out_b = (size_t)b * NK * H * W;
    const int NSLOT = TH * (W / 4);  // 2048, exactly 8 per thread
    for (int s = tid; s < NSLOT; s += NTHREADS) {
        const int yl = s >> 7;          // s / (W/4)
        const int x4 = (s & 127) << 2;  // 4-wide column start

        const int cbase = (yl + PAD) * LW + (x4 + PAD);
        const float c0 = tile[cbase + 0];
        const float c1 = tile[cbase + 1];
        const float c2 = tile[cbase + 2];
        const float c3 = tile[cbase + 3];

        const int nbase = yl * LW + x4;
        const size_t orow = out_b + (size_t)(y0 + yl) * W + x4;

        int k = 0;
#pragma unroll
        for (int i = 0; i < KSZ; ++i) {
#pragma unroll
            for (int j = 0; j < KSZ; ++j) {
                if (i == PAD && j == PAD) continue;
                const int nb = nbase + i * LW + j;  // i*LW+j folds to an imm offset
                v4f v;
                v.x = c0 - tile[nb + 0];
                v.y = c1 - tile[nb + 1];
                v.z = c2 - tile[nb + 2];
                v.w = c3 - tile[nb + 3];
                __builtin_nontemporal_store(
                    v, (v4f*)(out + (size_t)k * (H * W) + orow));
                ++k;
            }
        }
    }
}

extern "C" void kernel_launch(void* const* d_in, const int* in_sizes, int n_in,
                              void* d_out, int out_size, void* d_ws, size_t ws_size,
                              hipStream_t stream) {
    (void)in_sizes; (void)n_in; (void)d_ws; (void)ws_size; (void)out_size;
    const float* x = (const float*)d_in[0];
    float* out = (float*)d_out;
    dim3 grid(BATCH, H / TH);  // 16 x 32 = 512 blocks, 256 threads (8 waves)
    localvar_kernel<<<grid, NTHREADS, 0, stream>>>(x, out);
}